// MobileViTBlockv2_8881992368767
// MI455X (gfx1250) — compile-verified
//
#include <hip/hip_runtime.h>
#include <hip/hip_bf16.h>

typedef float v2f __attribute__((ext_vector_type(2)));
typedef float v8f __attribute__((ext_vector_type(8)));

#define EPSV 1e-5f

__device__ __forceinline__ v8f wmma4(v2f a, v2f b, v8f c) {
  // D(16x16,f32) = A(16x4,f32) * B(4x16,f32) + C
  return __builtin_amdgcn_wmma_f32_16x16x4_f32(false, a, false, b, (short)0, c,
                                               false, false);
}

// A fragment: row-major A, row includes (m0+lr), k2 = k + 2*lh -> contiguous pair
__device__ __forceinline__ v2f ldA(const float* A, int lda, int row, int k2) {
  return *(const v2f*)(A + (size_t)row * lda + k2);
}

// ---- LDS B-operand layout: K-pair interleaved + bank-skewed stride -------
// element (k, col) of a logical KxN matrix lives at
//   (k>>1)*(2N+32) + 2*col + (k&1)
// -> a (k2,k2+1) fragment is one contiguous 8B load; stride ≡ 32 (mod 64
//    banks) so upper half-wave (k2+2) uses the complementary banks.
__device__ __forceinline__ int bidx(int N, int k, int col) {
  return (k >> 1) * (2 * N + 32) + 2 * col + (k & 1);
}
__device__ __forceinline__ v2f ldB2(const float* Bm, int N, int col, int k2) {
  return *(const v2f*)(Bm + (size_t)(k2 >> 1) * (2 * N + 32) + 2 * col);
}

__device__ __forceinline__ float siluf(float x) {
  return x / (1.f + __expf(-x));
}

// ---------------------------------------------------------------- dw conv
__global__ __launch_bounds__(256) void dw_silu_kernel(
    const float* __restrict__ x, const float* __restrict__ w9,
    const float* __restrict__ dwb, const float* __restrict__ g,
    const float* __restrict__ bb, float* __restrict__ y) {
  int idx = blockIdx.x * 256 + threadIdx.x;              // total = 1<<25
  int wc = idx & 255;
  int h  = (idx >> 8) & 255;
  int c  = (idx >> 16) & 127;
  int b  = idx >> 23;
  const float* xp = x + ((size_t)(b * 128 + c) << 16);
  const float* wp = w9 + c * 9;
  float s = 0.f;
#pragma unroll
  for (int dy = -1; dy <= 1; ++dy) {
    int hh = h + dy;
    if ((unsigned)hh < 256u) {
#pragma unroll
      for (int dx = -1; dx <= 1; ++dx) {
        int ww = wc + dx;
        if ((unsigned)ww < 256u) s += xp[hh * 256 + ww] * wp[(dy + 1) * 3 + (dx + 1)];
      }
    }
  }
  s += dwb[c];
  s = s * (g[c] * rsqrtf(1.0f + EPSV)) + bb[c];
  y[idx] = siluf(s);
}

// ------------------------------------------------- pw_in GEMM -> patch layout
// p[b, d(256), m(65536)],  m = pp*1024 + n ; pixel = ((n>>5)*8 | pp>>3, (n&31)*8 | pp&7)
__global__ __launch_bounds__(256) void pwin_kernel(
    const float* __restrict__ y, const float* __restrict__ W,
    const float* __restrict__ bias, float* __restrict__ pbuf) {
  __shared__ __align__(16) float Xt[64 * (2 * 64 + 32)];  // K=128, N=64 -> 40KB
  int b = blockIdx.y;
  int m0 = blockIdx.x * 64;
  int tid = threadIdx.x;
  const float* yb = y + ((size_t)b << 23);
  {
    int j = tid & 63, r0 = tid >> 6;
    int m = m0 + j;
    int pp = m >> 10, n = m & 1023;
    int pix = ((((n >> 5) << 3) | (pp >> 3)) << 8) | (((n & 31) << 3) | (pp & 7));
    for (int r = r0; r < 128; r += 4)
      Xt[bidx(64, r, j)] = yb[((size_t)r << 16) + pix];
  }
  __syncthreads();
  int lane = tid & 31, wv = tid >> 5;
  int lh = lane >> 4, lr = lane & 15;
  int mrow = wv * 32;
  v8f acc[2][4];
#pragma unroll
  for (int s = 0; s < 2; ++s)
#pragma unroll
    for (int t = 0; t < 4; ++t)
#pragma unroll
      for (int e = 0; e < 8; ++e) acc[s][t][e] = 0.f;
  for (int k = 0; k < 128; k += 4) {
    int k2 = k + 2 * lh;
    v2f a0 = ldA(W, 128, mrow + lr, k2);
    v2f a1 = ldA(W, 128, mrow + 16 + lr, k2);
#pragma unroll
    for (int t = 0; t < 4; ++t) {
      v2f bv = ldB2(Xt, 64, t * 16 + lr, k2);
      acc[0][t] = wmma4(a0, bv, acc[0][t]);
      acc[1][t] = wmma4(a1, bv, acc[1][t]);
    }
  }
  float* pb = pbuf + ((size_t)b << 24);
#pragma unroll
  for (int s = 0; s < 2; ++s)
#pragma unroll
    for (int t = 0; t < 4; ++t)
#pragma unroll
      for (int e = 0; e < 8; ++e) {
        int row = mrow + s * 16 + e + 8 * lh;
        int col = t * 16 + lr;
        pb[((size_t)row << 16) + m0 + col] = acc[s][t][e] + bias[row];
      }
}

// ---------------------------------------------------------------- GN stats
__global__ void zero_stats_kernel(float* stats) {
  if (threadIdx.x < 8) stats[threadIdx.x] = 0.f;
}

__global__ __launch_bounds__(256) void gnstats_kernel(
    const float* __restrict__ p, float* __restrict__ stats) {
  __shared__ float ls[256], lq[256];
  int b = blockIdx.y, tid = threadIdx.x;
  const float* pb = p + ((size_t)b << 24);
  float s = 0.f, q = 0.f;
  for (size_t i = (size_t)blockIdx.x * 256 + tid; i < ((size_t)1 << 24);
       i += (size_t)gridDim.x * 256) {
    float v = pb[i];
    s += v; q += v * v;
  }
  ls[tid] = s; lq[tid] = q;
  __syncthreads();
  for (int o = 128; o > 0; o >>= 1) {
    if (tid < o) { ls[tid] += ls[tid + o]; lq[tid] += lq[tid + o]; }
    __syncthreads();
  }
  if (tid == 0) {
    atomicAdd(&stats[b * 2], ls[0]);
    atomicAdd(&stats[b * 2 + 1], lq[0]);
  }
}

__global__ void finalize_kernel(const float* stats, float* mr) {
  int b = threadIdx.x;
  if (b < 4) {
    const float n = 16777216.f;  // 256*64*1024
    float m = stats[b * 2] / n;
    float v = stats[b * 2 + 1] / n - m * m;
    mr[b * 2] = m;
    mr[b * 2 + 1] = rsqrtf(v + EPSV);
  }
}

// -------------------------------- fold GN (and optional bn2) into W and bias
// Wf[b,o,c] = W[o,c]*g[c]*r_b*rs(o) ;  bf[b,o] = (bias[o] + W@(bt - m r g))*rs + ra
__global__ __launch_bounds__(256) void fold_kernel(
    const float* __restrict__ W, const float* __restrict__ bias,
    const float* __restrict__ g, const float* __restrict__ bt,
    const float* __restrict__ mr, float* __restrict__ Wf, float* __restrict__ bf,
    const float* __restrict__ rsg, const float* __restrict__ ra, int O) {
  __shared__ float red[256];
  int b = blockIdx.y, o = blockIdx.x, c = threadIdx.x;
  float m = mr[b * 2], r = mr[b * 2 + 1];
  float w = W[(size_t)o * 256 + c];
  float gg = g[c] * r;
  float rs = rsg ? (rsg[o] * rsqrtf(1.f + EPSV)) : 1.f;
  Wf[((size_t)(b * O + o)) * 256 + c] = w * gg * rs;
  red[c] = w * (bt[c] - m * gg);
  __syncthreads();
  for (int off = 128; off > 0; off >>= 1) {
    if (c < off) red[c] += red[c + off];
    __syncthreads();
  }
  if (c == 0) bf[b * O + o] = (bias[o] + red[0]) * rs + (ra ? ra[o] : 0.f);
}

// ------------------------------------------------ q projection + softmax(N)
__global__ __launch_bounds__(256) void qsoftmax_kernel(
    const float* __restrict__ pbuf, const float* __restrict__ wfold,
    const float* __restrict__ bfold, float* __restrict__ scores) {
  __shared__ float wq[256];
  __shared__ float red[256];
  int b = blockIdx.y, pp = blockIdx.x, tid = threadIdx.x;
  wq[tid] = wfold[(size_t)b * 513 * 256 + tid];  // row 0 (q)
  __syncthreads();
  const float* pb = pbuf + ((size_t)b << 24) + ((size_t)pp << 10);
  float bq = bfold[b * 513];
  float q0 = bq, q1 = bq, q2 = bq, q3 = bq;
  for (int c = 0; c < 256; ++c) {
    const float* row = pb + ((size_t)c << 16);
    float wc = wq[c];
    q0 += wc * row[tid];
    q1 += wc * row[tid + 256];
    q2 += wc * row[tid + 512];
    q3 += wc * row[tid + 768];
  }
  float mx = fmaxf(fmaxf(q0, q1), fmaxf(q2, q3));
  red[tid] = mx; __syncthreads();
  for (int o = 128; o > 0; o >>= 1) {
    if (tid < o) red[tid] = fmaxf(red[tid], red[tid + o]);
    __syncthreads();
  }
  mx = red[0]; __syncthreads();
  float e0 = __expf(q0 - mx), e1 = __expf(q1 - mx);
  float e2 = __expf(q2 - mx), e3 = __expf(q3 - mx);
  red[tid] = e0 + e1 + e2 + e3; __syncthreads();
  for (int o = 128; o > 0; o >>= 1) {
    if (tid < o) red[tid] += red[tid + o];
    __syncthreads();
  }
  float inv = 1.f / red[0];
  float* sc = scores + ((size_t)(b * 64 + pp) << 10);
  sc[tid] = e0 * inv; sc[tid + 256] = e1 * inv;
  sc[tid + 512] = e2 * inv; sc[tid + 768] = e3 * inv;
}

// -------------------------------------------- zbar[b,c,pp] = sum_n s[n]*p[c,n]
__global__ __launch_bounds__(128) void zbar_kernel(
    const float* __restrict__ pbuf, const float* __restrict__ scores,
    float* __restrict__ zb) {
  __shared__ float red[128];
  int pp = blockIdx.x, c = blockIdx.y, b = blockIdx.z, tid = threadIdx.x;
  const float* row = pbuf + ((size_t)b << 24) + ((size_t)c << 16) + ((size_t)pp << 10);
  const float* sc = scores + ((size_t)(b * 64 + pp) << 10);
  float s = 0.f;
  for (int n = tid; n < 1024; n += 128) s += row[n] * sc[n];
  red[tid] = s; __syncthreads();
  for (int o = 64; o > 0; o >>= 1) {
    if (tid < o) red[tid] += red[tid + o];
    __syncthreads();
  }
  if (tid == 0) zb[(size_t)(b * 256 + c) * 64 + pp] = red[0];
}

// ------------------------------------- ctx[b,d,pp] = Wk'[d,:]@zbar[:,pp] + bk'
__global__ __launch_bounds__(64) void ctx_kernel(
    const float* __restrict__ wfold, const float* __restrict__ bfold,
    const float* __restrict__ zb, float* __restrict__ ctxb) {
  int d = blockIdx.x, b = blockIdx.y, pp = threadIdx.x;
  const float* Wk = wfold + ((size_t)b * 513 + 1 + d) * 256;
  const float* zbb = zb + (size_t)b * 256 * 64;
  float s = bfold[b * 513 + 1 + d];
  for (int c = 0; c < 256; ++c) s += Wk[c] * zbb[c * 64 + pp];
  ctxb[(size_t)(b * 256 + d) * 64 + pp] = s;
}

// ------- fused: V=Wv'@P+bv' -> relu*ctx -> out_w@A + out_b -> residual into p
// 32-column tiles (all share one within-patch position pp)
__global__ __launch_bounds__(256) void attn_out_kernel(
    float* __restrict__ pbuf, const float* __restrict__ wfold,
    const float* __restrict__ bfold, const float* __restrict__ ctxb,
    const float* __restrict__ ow, const float* __restrict__ ob) {
  __shared__ __align__(16) float Pt[128 * (2 * 32 + 32)];  // K=256,N=32 -> 48KB
  __shared__ __align__(16) float At[128 * (2 * 32 + 32)];  // K=256,N=32 -> 48KB
  __shared__ float ctxs[256];
  int b = blockIdx.y;
  int m0 = blockIdx.x * 32;
  int pp = m0 >> 10;
  int tid = threadIdx.x;
  float* pb = pbuf + ((size_t)b << 24);
  {
    int j = tid & 31, r0 = tid >> 5;
    for (int r = r0; r < 256; r += 8)
      Pt[bidx(32, r, j)] = pb[((size_t)r << 16) + m0 + j];
  }
  ctxs[tid] = ctxb[(size_t)(b * 256 + tid) * 64 + pp];
  __syncthreads();
  int lane = tid & 31, wv = tid >> 5;
  int lh = lane >> 4, lr = lane & 15;
  int mrow = wv * 32;
  {  // phase 1: V rows (folded qkv rows 257..512)
    const float* A = wfold + ((size_t)b * 513 + 257) * 256;
    const float* bia = bfold + b * 513 + 257;
    v8f acc[2][2];
#pragma unroll
    for (int s = 0; s < 2; ++s)
#pragma unroll
      for (int t = 0; t < 2; ++t)
#pragma unroll
        for (int e = 0; e < 8; ++e) acc[s][t][e] = 0.f;
    for (int k = 0; k < 256; k += 4) {
      int k2 = k + 2 * lh;
      v2f a0 = ldA(A, 256, mrow + lr, k2);
      v2f a1 = ldA(A, 256, mrow + 16 + lr, k2);
      v2f b0 = ldB2(Pt, 32, lr, k2);
      v2f b1 = ldB2(Pt, 32, 16 + lr, k2);
      acc[0][0] = wmma4(a0, b0, acc[0][0]);
      acc[0][1] = wmma4(a0, b1, acc[0][1]);
      acc[1][0] = wmma4(a1, b0, acc[1][0]);
      acc[1][1] = wmma4(a1, b1, acc[1][1]);
    }
#pragma unroll
    for (int s = 0; s < 2; ++s)
#pragma unroll
      for (int t = 0; t < 2; ++t)
#pragma unroll
        for (int e = 0; e < 8; ++e) {
          int row = mrow + s * 16 + e + 8 * lh;
          int col = t * 16 + lr;
          float v = acc[s][t][e] + bia[row];
          At[bidx(32, row, col)] = fmaxf(v, 0.f) * ctxs[row];
        }
  }
  __syncthreads();
  {  // phase 2: delta = out_w @ A + out_b ; p += delta (residual via Pt)
    v8f acc[2][2];
#pragma unroll
    for (int s = 0; s < 2; ++s)
#pragma unroll
      for (int t = 0; t < 2; ++t)
#pragma unroll
        for (int e = 0; e < 8; ++e) acc[s][t][e] = 0.f;
    for (int k = 0; k < 256; k += 4) {
      int k2 = k + 2 * lh;
      v2f a0 = ldA(ow, 256, mrow + lr, k2);
      v2f a1 = ldA(ow, 256, mrow + 16 + lr, k2);
      v2f b0 = ldB2(At, 32, lr, k2);
      v2f b1 = ldB2(At, 32, 16 + lr, k2);
      acc[0][0] = wmma4(a0, b0, acc[0][0]);
      acc[0][1] = wmma4(a0, b1, acc[0][1]);
      acc[1][0] = wmma4(a1, b0, acc[1][0]);
      acc[1][1] = wmma4(a1, b1, acc[1][1]);
    }
#pragma unroll
    for (int s = 0; s < 2; ++s)
#pragma unroll
      for (int t = 0; t < 2; ++t)
#pragma unroll
        for (int e = 0; e < 8; ++e) {
          int row = mrow + s * 16 + e + 8 * lh;
          int col = t * 16 + lr;
          pb[((size_t)row << 16) + m0 + col] =
              Pt[bidx(32, row, col)] + acc[s][t][e] + ob[row];
        }
  }
}

// ---------- fused FFN: h = silu(Wf1'@P + bf1') ; p += ffn2_w@h + ffn2_b
__global__ __launch_bounds__(256) void ffn_kernel(
    float* __restrict__ pbuf, const float* __restrict__ wf1,
    const float* __restrict__ bf1, const float* __restrict__ w2,
    const float* __restrict__ b2) {
  __shared__ __align__(16) float Pt[128 * (2 * 32 + 32)];  // K=256,N=32 -> 48KB
  __shared__ __align__(16) float Ht[256 * (2 * 32 + 32)];  // K=512,N=32 -> 96KB
  int b = blockIdx.y;
  int m0 = blockIdx.x * 32;
  int tid = threadIdx.x;
  float* pb = pbuf + ((size_t)b << 24);
  {
    int j = tid & 31, r0 = tid >> 5;
    for (int r = r0; r < 256; r += 8)
      Pt[bidx(32, r, j)] = pb[((size_t)r << 16) + m0 + j];
  }
  __syncthreads();
  int lane = tid & 31, wv = tid >> 5;
  int lh = lane >> 4, lr = lane & 15;
  {  // phase 1: 512 rows, each wave 64 rows x 32 cols
    const float* A = wf1 + ((size_t)b << 17);
    const float* bia = bf1 + (b << 9);
    int mrow = wv * 64;
    v8f acc[4][2];
#pragma unroll
    for (int s = 0; s < 4; ++s)
#pragma unroll
      for (int t = 0; t < 2; ++t)
#pragma unroll
        for (int e = 0; e < 8; ++e) acc[s][t][e] = 0.f;
    for (int k = 0; k < 256; k += 4) {
      int k2 = k + 2 * lh;
      v2f a[4];
#pragma unroll
      for (int s = 0; s < 4; ++s) a[s] = ldA(A, 256, mrow + s * 16 + lr, k2);
      v2f b0 = ldB2(Pt, 32, lr, k2);
      v2f b1 = ldB2(Pt, 32, 16 + lr, k2);
#pragma unroll
      for (int s = 0; s < 4; ++s) {
        acc[s][0] = wmma4(a[s], b0, acc[s][0]);
        acc[s][1] = wmma4(a[s], b1, acc[s][1]);
      }
    }
#pragma unroll
    for (int s = 0; s < 4; ++s)
#pragma unroll
      for (int t = 0; t < 2; ++t)
#pragma unroll
        for (int e = 0; e < 8; ++e) {
          int row = mrow + s * 16 + e + 8 * lh;
          int col = t * 16 + lr;
          float h = acc[s][t][e] + bia[row];
          Ht[bidx(32, row, col)] = siluf(h);
        }
  }
  __syncthreads();
  {  // phase 2: 256 rows, K=512
    int mrow = wv * 32;
    v8f acc[2][2];
#pragma unroll
    for (int s = 0; s < 2; ++s)
#pragma unroll
      for (int t = 0; t < 2; ++t)
#pragma unroll
        for (int e = 0; e < 8; ++e) acc[s][t][e] = 0.f;
    for (int k = 0; k < 512; k += 4) {
      int k2 = k + 2 * lh;
      v2f a0 = ldA(w2, 512, mrow + lr, k2);
      v2f a1 = ldA(w2, 512, mrow + 16 + lr, k2);
      v2f b0 = ldB2(Ht, 32, lr, k2);
      v2f b1 = ldB2(Ht, 32, 16 + lr, k2);
      acc[0][0] = wmma4(a0, b0, acc[0][0]);
      acc[0][1] = wmma4(a0, b1, acc[0][1]);
      acc[1][0] = wmma4(a1, b0, acc[1][0]);
      acc[1][1] = wmma4(a1, b1, acc[1][1]);
    }
#pragma unroll
    for (int s = 0; s < 2; ++s)
#pragma unroll
      for (int t = 0; t < 2; ++t)
#pragma unroll
        for (int e = 0; e < 8; ++e) {
          int row = mrow + s * 16 + e + 8 * lh;
          int col = t * 16 + lr;
          pb[((size_t)row << 16) + m0 + col] =
              Pt[bidx(32, row, col)] + acc[s][t][e] + b2[row];
        }
  }
}

// ------- final: out = (proj@gnf(p))*bn2 fold, scattered back to pixel layout
__global__ __launch_bounds__(256) void proj_kernel(
    const float* __restrict__ pbuf, const float* __restrict__ Wp,
    const float* __restrict__ bp, float* __restrict__ out) {
  __shared__ __align__(16) float Pt[128 * (2 * 64 + 32)];  // K=256,N=64 -> 80KB
  int b = blockIdx.y;
  int m0 = blockIdx.x * 64;
  int tid = threadIdx.x;
  const float* pb = pbuf + ((size_t)b << 24);
  {
    int j = tid & 63, r0 = tid >> 6;
    for (int r = r0; r < 256; r += 4)
      Pt[bidx(64, r, j)] = pb[((size_t)r << 16) + m0 + j];
  }
  __syncthreads();
  int lane = tid & 31, wv = tid >> 5;
  int lh = lane >> 4, lr = lane & 15;
  int mrow = (wv >> 1) * 32;
  int nc0 = (wv & 1) * 32;
  const float* A = Wp + ((size_t)b << 15);
  const float* bia = bp + (b << 7);
  v8f acc[2][2];
#pragma unroll
  for (int s = 0; s < 2; ++s)
#pragma unroll
    for (int t = 0; t < 2; ++t)
#pragma unroll
      for (int e = 0; e < 8; ++e) acc[s][t][e] = 0.f;
  for (int k = 0; k < 256; k += 4) {
    int k2 = k + 2 * lh;
    v2f a0 = ldA(A, 256, mrow + lr, k2);
    v2f a1 = ldA(A, 256, mrow + 16 + lr, k2);
    v2f b0 = ldB2(Pt, 64, nc0 + lr, k2);
    v2f b1 = ldB2(Pt, 64, nc0 + 16 + lr, k2);
    acc[0][0] = wmma4(a0, b0, acc[0][0]);
    acc[0][1] = wmma4(a0, b1, acc[0][1]);
    acc[1][0] = wmma4(a1, b0, acc[1][0]);
    acc[1][1] = wmma4(a1, b1, acc[1][1]);
  }
  float* ob = out + ((size_t)b << 23);
#pragma unroll
  for (int s = 0; s < 2; ++s)
#pragma unroll
    for (int t = 0; t < 2; ++t)
#pragma unroll
      for (int e = 0; e < 8; ++e) {
        int row = mrow + s * 16 + e + 8 * lh;
        int col = nc0 + t * 16 + lr;
        int m = m0 + col;
        int pq = m >> 10, n = m & 1023;
        int pix = ((((n >> 5) << 3) | (pq >> 3)) << 8) | (((n & 31) << 3) | (pq & 7));
        ob[((size_t)row << 16) + pix] = acc[s][t][e] + bia[row];
      }
}

// ============================================================== launcher
extern "C" void kernel_launch(void* const* d_in, const int* in_sizes, int n_in,
                              void* d_out, int out_size, void* d_ws, size_t ws_size,
                              hipStream_t stream) {
  (void)in_sizes; (void)n_in; (void)out_size; (void)ws_size;
  const float* x      = (const float*)d_in[0];
  const float* dw_w   = (const float*)d_in[1];
  const float* dw_b   = (const float*)d_in[2];
  const float* bn1_g  = (const float*)d_in[3];
  const float* bn1_b  = (const float*)d_in[4];
  const float* pw_in_w= (const float*)d_in[5];
  const float* pw_in_b= (const float*)d_in[6];
  const float* gn1_g  = (const float*)d_in[7];
  const float* gn1_b  = (const float*)d_in[8];
  const float* qkv_w  = (const float*)d_in[9];
  const float* qkv_b  = (const float*)d_in[10];
  const float* out_w  = (const float*)d_in[11];
  const float* out_b  = (const float*)d_in[12];
  const float* gn2_g  = (const float*)d_in[13];
  const float* gn2_b  = (const float*)d_in[14];
  const float* ffn1_w = (const float*)d_in[15];
  const float* ffn1_b = (const float*)d_in[16];
  const float* ffn2_w = (const float*)d_in[17];
  const float* ffn2_b = (const float*)d_in[18];
  const float* gnf_g  = (const float*)d_in[19];
  const float* gnf_b  = (const float*)d_in[20];
  const float* proj_w = (const float*)d_in[21];
  const float* proj_b = (const float*)d_in[22];
  const float* bn2_g  = (const float*)d_in[23];
  const float* bn2_b  = (const float*)d_in[24];

  float* ws = (float*)d_ws;
  size_t off = 0;
  float* yb     = ws + off; off += (size_t)4 * 128 * 65536;   // 128 MB
  float* pbuf   = ws + off; off += (size_t)4 * 256 * 65536;   // 256 MB
  float* scores = ws + off; off += (size_t)4 * 64 * 1024;
  float* zb     = ws + off; off += (size_t)4 * 256 * 64;
  float* ctxb   = ws + off; off += (size_t)4 * 256 * 64;
  float* wfold  = ws + off; off += (size_t)4 * 513 * 256;
  float* bfold  = ws + off; off += (size_t)4 * 513;
  float* wf1    = ws + off; off += (size_t)4 * 512 * 256;
  float* bf1    = ws + off; off += (size_t)4 * 512;
  float* wpr    = ws + off; off += (size_t)4 * 128 * 256;
  float* bpr    = ws + off; off += (size_t)4 * 128;
  float* stats  = ws + off; off += 8;
  float* mr     = ws + off; off += 8;

  dw_silu_kernel<<<(1 << 25) / 256, 256, 0, stream>>>(x, dw_w, dw_b, bn1_g, bn1_b, yb);
  pwin_kernel<<<dim3(1024, 4), 256, 0, stream>>>(yb, pw_in_w, pw_in_b, pbuf);

  for (int i = 0; i < 2; ++i) {
    zero_stats_kernel<<<1, 32, 0, stream>>>(stats);
    gnstats_kernel<<<dim3(2048, 4), 256, 0, stream>>>(pbuf, stats);
    finalize_kernel<<<1, 32, 0, stream>>>(stats, mr);
    fold_kernel<<<dim3(513, 4), 256, 0, stream>>>(
        qkv_w + (size_t)i * 513 * 256, qkv_b + i * 513,
        gn1_g + i * 256, gn1_b + i * 256, mr, wfold, bfold, nullptr, nullptr, 513);
    qsoftmax_kernel<<<dim3(64, 4), 256, 0, stream>>>(pbuf, wfold, bfold, scores);
    zbar_kernel<<<dim3(64, 256, 4), 128, 0, stream>>>(pbuf, scores, zb);
    ctx_kernel<<<dim3(256, 4), 64, 0, stream>>>(wfold, bfold, zb, ctxb);
    attn_out_kernel<<<dim3(2048, 4), 256, 0, stream>>>(
        pbuf, wfold, bfold, ctxb, out_w + (size_t)i * 256 * 256, out_b + i * 256);

    zero_stats_kernel<<<1, 32, 0, stream>>>(stats);
    gnstats_kernel<<<dim3(2048, 4), 256, 0, stream>>>(pbuf, stats);
    finalize_kernel<<<1, 32, 0, stream>>>(stats, mr);
    fold_kernel<<<dim3(512, 4), 256, 0, stream>>>(
        ffn1_w + (size_t)i * 512 * 256, ffn1_b + i * 512,
        gn2_g + i * 256, gn2_b + i * 256, mr, wf1, bf1, nullptr, nullptr, 512);
    ffn_kernel<<<dim3(2048, 4), 256, 0, stream>>>(
        pbuf, wf1, bf1, ffn2_w + (size_t)i * 256 * 512, ffn2_b + i * 256);
  }

  zero_stats_kernel<<<1, 32, 0, stream>>>(stats);
  gnstats_kernel<<<dim3(2048, 4), 256, 0, stream>>>(pbuf, stats);
  finalize_kernel<<<1, 32, 0, stream>>>(stats, mr);
  fold_kernel<<<dim3(128, 4), 256, 0, stream>>>(
      proj_w, proj_b, gnf_g, gnf_b, mr, wpr, bpr, bn2_g, bn2_b, 128);
  proj_kernel<<<dim3(1024, 4), 256, 0, stream>>>(pbuf, wpr, bpr, (float*)d_out);
}